// UniMP_21131239096636
// MI455X (gfx1250) — compile-verified
//
#include <hip/hip_runtime.h>
#include <hip/hip_bf16.h>
#include <math.h>

// Problem dims (all divide 16 exactly -> no WMMA tile guards, EXEC stays all-1s)
#define N_NODES 50000
#define N_EDGES 800000
#define N_GRAPH 512
#define D       64
#define DE      16
#define LDS_STRIDE 68   // padded row stride (floats): bank = (4l+ka)%64 -> conflict-free

typedef __attribute__((ext_vector_type(2))) float v2f;
typedef __attribute__((ext_vector_type(8))) float v8f;
typedef __attribute__((ext_vector_type(2))) int   v2i;

// address-space qualified v2i for the async-to-LDS builtin's typed pointers
typedef __attribute__((address_space(1))) v2i gv2i;   // global
typedef __attribute__((address_space(3))) v2i lv2i;   // LDS

__device__ __forceinline__ float leakyf(float z) { return z >= 0.f ? z : 0.01f * z; }

// float atomic-max via monotone integer mapping (sign-split trick)
__device__ __forceinline__ void atomicMaxFloat(float* addr, float val) {
    if (val >= 0.0f) atomicMax((int*)addr, __float_as_int(val));
    else             atomicMin((unsigned int*)addr, __float_as_uint(val));
}

__global__ void fill_kernel(float* p, float v, int n) {
    int i = blockIdx.x * blockDim.x + threadIdx.x;
    if (i < n) p[i] = v;
}

// Fused 4-way projection: {Y0..Y3}[16 rows x 64] = A_tile @ {W0..W3} + {B0..B3}
// Block = 512 threads = 16 waves, one block per 16-row tile of A.
// A tile staged once into LDS (async-to-LDS when available), each wave computes
// one (matrix, col-tile) 16x16 WMMA output with V_WMMA_F32_16X16X4_F32.
__global__ void __launch_bounds__(512)
proj4_wmma(const float* __restrict__ A,
           const float* __restrict__ W0, const float* __restrict__ W1,
           const float* __restrict__ W2, const float* __restrict__ W3,
           const float* __restrict__ B0, const float* __restrict__ B1,
           const float* __restrict__ B2, const float* __restrict__ B3,
           float* __restrict__ Y0, float* __restrict__ Y1,
           float* __restrict__ Y2, float* __restrict__ Y3) {
    __shared__ float ldsX[16 * LDS_STRIDE];
    const int t    = threadIdx.x;
    const int row0 = blockIdx.x * 16;

    // ---- stage 16x64 A tile, fully coalesced (2 floats / thread) ----
    {
        const int row = t >> 5;            // 16 rows, 32 float-pairs per row
        const int cp  = (t & 31) * 2;
        const float* g = A + (size_t)(row0 + row) * D + cp;
        float*       l = &ldsX[row * LDS_STRIDE + cp];
#if __has_builtin(__builtin_amdgcn_global_load_async_to_lds_b64)
        __builtin_amdgcn_global_load_async_to_lds_b64(
            (gv2i*)g, (lv2i*)l, 0, 0);
#if __has_builtin(__builtin_amdgcn_s_wait_asynccnt)
        __builtin_amdgcn_s_wait_asynccnt(0);
#endif
#else
        l[0] = g[0];
        l[1] = g[1];
#endif
    }
    __syncthreads();

    // ---- per-wave WMMA: wave w -> matrix (w>>2), column tile (w&3) ----
    const int w    = t >> 5;
    const int mi   = w >> 2;
    const int col0 = (w & 3) * 16;
    const int lane = t & 31;
    const int half = lane >> 4;            // 0: lanes 0-15, 1: lanes 16-31
    const int l    = lane & 15;
    const float* Wm = (mi == 0) ? W0 : (mi == 1) ? W1 : (mi == 2) ? W2 : W3;
    const float* Bm = (mi == 0) ? B0 : (mi == 1) ? B1 : (mi == 2) ? B2 : B3;
    float*       Ym = (mi == 0) ? Y0 : (mi == 1) ? Y1 : (mi == 2) ? Y2 : Y3;

    v8f acc = {};
    for (int kk = 0; kk < D; kk += 4) {
        const int ka = kk + half * 2;      // A: lanes 0-15 K=kk,kk+1 ; 16-31 K=kk+2,kk+3
        v2f a, b;
        a.x = ldsX[l * LDS_STRIDE + ka];
        a.y = ldsX[l * LDS_STRIDE + ka + 1];
        b.x = Wm[(size_t)(ka)     * D + col0 + l];
        b.y = Wm[(size_t)(ka + 1) * D + col0 + l];
        acc = __builtin_amdgcn_wmma_f32_16x16x4_f32(false, a, false, b,
                                                    (short)0, acc, false, false);
    }
    const float bv = Bm[col0 + l];
    for (int r = 0; r < 8; ++r) {          // C/D: VGPR r -> M=r (+8 for upper lanes)
        Ym[(size_t)(row0 + r + 8 * half) * D + col0 + l] = acc[r] + bv;
    }
}

// One wave per edge: score[e] = dot(q[dst], k[src] + ea[e]@We) / sqrt(64)
__global__ void edge_score_kernel(const float* __restrict__ q, const float* __restrict__ k,
                                  const float* __restrict__ ea, const float* __restrict__ we,
                                  const int* __restrict__ src, const int* __restrict__ dst,
                                  float* __restrict__ score, float* __restrict__ m) {
    int e = blockIdx.x * (blockDim.x >> 5) + (threadIdx.x >> 5);
    if (e >= N_EDGES) return;
    const int lane = threadIdx.x & 31;
    if (e + 256 < N_EDGES)                 // prefetch the streamed edge_attr ahead
        __builtin_prefetch(ea + (size_t)(e + 256) * DE, 0, 1);
    const int s = src[e], d = dst[e];
    const int c0 = lane, c1 = lane + 32;
    float ee0 = 0.f, ee1 = 0.f;
    #pragma unroll
    for (int j = 0; j < DE; ++j) {
        float a = ea[(size_t)e * DE + j];
        ee0 = fmaf(a, we[j * D + c0], ee0);
        ee1 = fmaf(a, we[j * D + c1], ee1);
    }
    float p = q[(size_t)d * D + c0] * (k[(size_t)s * D + c0] + ee0)
            + q[(size_t)d * D + c1] * (k[(size_t)s * D + c1] + ee1);
    #pragma unroll
    for (int off = 16; off > 0; off >>= 1) p += __shfl_xor(p, off, 32);
    if (lane == 0) {
        float sc = p * 0.125f;             // 1/sqrt(64)
        score[e] = sc;
        atomicMaxFloat(&m[d], sc);
    }
}

// One thread per edge: a = exp(score - m[dst]); den[dst] += a
__global__ void edge_exp_kernel(float* __restrict__ score, const float* __restrict__ m,
                                const int* __restrict__ dst, float* __restrict__ den) {
    int e = blockIdx.x * blockDim.x + threadIdx.x;
    if (e >= N_EDGES) return;
    int d = dst[e];
    float a = expf(score[e] - m[d]);
    score[e] = a;
    atomicAdd(&den[d], a);
}

// One wave per edge: H[dst] += (a/den[dst]) * (v[src] + ea[e]@We)
__global__ void edge_agg_kernel(const float* __restrict__ v, const float* __restrict__ ea,
                                const float* __restrict__ we, const float* __restrict__ score,
                                const float* __restrict__ den,
                                const int* __restrict__ src, const int* __restrict__ dst,
                                float* __restrict__ H) {
    int e = blockIdx.x * (blockDim.x >> 5) + (threadIdx.x >> 5);
    if (e >= N_EDGES) return;
    const int lane = threadIdx.x & 31;
    if (e + 256 < N_EDGES)
        __builtin_prefetch(ea + (size_t)(e + 256) * DE, 0, 1);
    const int s = src[e], d = dst[e];
    float dn = den[d];
    float alpha = score[e] / (dn == 0.f ? 1.f : dn);
    const int c0 = lane, c1 = lane + 32;
    float ee0 = 0.f, ee1 = 0.f;
    #pragma unroll
    for (int j = 0; j < DE; ++j) {
        float a = ea[(size_t)e * DE + j];
        ee0 = fmaf(a, we[j * D + c0], ee0);
        ee1 = fmaf(a, we[j * D + c1], ee1);
    }
    atomicAdd(&H[(size_t)d * D + c0], alpha * (v[(size_t)s * D + c0] + ee0));
    atomicAdd(&H[(size_t)d * D + c1], alpha * (v[(size_t)s * D + c1] + ee1));
}

__global__ void leaky_kernel(float* p, int n) {
    int i = blockIdx.x * blockDim.x + threadIdx.x;
    if (i < n) p[i] = leakyf(p[i]);
}

// pooled[batch[n]] += emb[n]  (pooled = 128 KB, lives in L2)
__global__ void pool_kernel(const float* __restrict__ emb, const int* __restrict__ batch,
                            float* __restrict__ pooled) {
    int i = blockIdx.x * blockDim.x + threadIdx.x;
    if (i >= N_NODES * D) return;
    int node = i >> 6, c = i & 63;
    atomicAdd(&pooled[(size_t)batch[node] * D + c], emb[i]);
}

// Per-graph head: L2-normalize -> fc1 (64x64) -> leaky -> fc2 (64x1). Block = 64 thr.
__global__ void head_kernel(const float* __restrict__ pooled,
                            const float* __restrict__ w1, const float* __restrict__ b1,
                            const float* __restrict__ w2, const float* __restrict__ b2,
                            float* __restrict__ out) {
    __shared__ float h[D];
    __shared__ float red[D];
    const int g = blockIdx.x, t = threadIdx.x;
    float p = pooled[(size_t)g * D + t];
    red[t] = p * p;
    __syncthreads();
    for (int s = 32; s > 0; s >>= 1) { if (t < s) red[t] += red[t + s]; __syncthreads(); }
    float inv = 1.0f / fmaxf(sqrtf(red[0]), 1e-12f);
    h[t] = p * inv;
    __syncthreads();                       // also fences the red[0] read above
    float acc = b1[t];
    #pragma unroll 8
    for (int i = 0; i < D; ++i) acc = fmaf(h[i], w1[i * D + t], acc);
    acc = leakyf(acc);
    red[t] = acc * w2[t];
    __syncthreads();
    for (int s = 32; s > 0; s >>= 1) { if (t < s) red[t] += red[t + s]; __syncthreads(); }
    if (t == 0) out[g] = red[0] + b2[0];
}

extern "C" void kernel_launch(void* const* d_in, const int* in_sizes, int n_in,
                              void* d_out, int out_size, void* d_ws, size_t ws_size,
                              hipStream_t stream) {
    const float* x     = (const float*)d_in[0];
    const int*   ei    = (const int*)d_in[1];     // [2,E]: row 0 = src, row 1 = dst
    const float* ea    = (const float*)d_in[2];
    const int*   batch = (const int*)d_in[3];
    const float *wq1 = (const float*)d_in[4],  *bq1 = (const float*)d_in[5];
    const float *wk1 = (const float*)d_in[6],  *bk1 = (const float*)d_in[7];
    const float *wv1 = (const float*)d_in[8],  *bv1 = (const float*)d_in[9];
    const float *we1 = (const float*)d_in[10];
    const float *ws1 = (const float*)d_in[11], *bs1 = (const float*)d_in[12];
    const float *wq2 = (const float*)d_in[13], *bq2 = (const float*)d_in[14];
    const float *wk2 = (const float*)d_in[15], *bk2 = (const float*)d_in[16];
    const float *wv2 = (const float*)d_in[17], *bv2 = (const float*)d_in[18];
    const float *we2 = (const float*)d_in[19];
    const float *ws2 = (const float*)d_in[20], *bs2 = (const float*)d_in[21];
    const float *wf1 = (const float*)d_in[22], *bf1 = (const float*)d_in[23];
    const float *wf2 = (const float*)d_in[24], *bf2 = (const float*)d_in[25];

    const int* src = ei;
    const int* dst = ei + N_EDGES;

    // Workspace layout (~55 MB of float scratch)
    float* ws     = (float*)d_ws;
    float* q      = ws;
    float* k      = q  + (size_t)N_NODES * D;
    float* v      = k  + (size_t)N_NODES * D;
    float* h1     = v  + (size_t)N_NODES * D;    // layer-1 skip+agg / layer-2 input
    float* sc     = h1 + (size_t)N_NODES * D;    // E  (score -> exp(score-m))
    float* mbuf   = sc + N_EDGES;                // N  (segment max)
    float* den    = mbuf + N_NODES;              // N  (segment sum)
    float* pooled = den + N_NODES;               // G*64

    float* outv = (float*)d_out;                 // [G]  graph outputs
    float* h2   = outv + N_GRAPH;                // [N,64] atom_embs, built in place

    const int nTiles  = N_NODES / 16;            // 3125 row tiles
    const int nThrN   = (N_NODES + 255) / 256;
    const int nThrNE  = (N_NODES * D + 255) / 256;
    const int edgeBlk = N_EDGES / 8;             // 8 waves (edges) per 256-thr block
    const int edgeThr = (N_EDGES + 255) / 256;

    // ---------------- Layer 1 (input x -> h1) ----------------
    fill_kernel<<<nThrN, 256, 0, stream>>>(mbuf, -INFINITY, N_NODES);
    fill_kernel<<<nThrN, 256, 0, stream>>>(den, 0.f, N_NODES);
    proj4_wmma<<<nTiles, 512, 0, stream>>>(x, wq1, wk1, wv1, ws1,
                                           bq1, bk1, bv1, bs1, q, k, v, h1);
    edge_score_kernel<<<edgeBlk, 256, 0, stream>>>(q, k, ea, we1, src, dst, sc, mbuf);
    edge_exp_kernel<<<edgeThr, 256, 0, stream>>>(sc, mbuf, dst, den);
    edge_agg_kernel<<<edgeBlk, 256, 0, stream>>>(v, ea, we1, sc, den, src, dst, h1);
    leaky_kernel<<<nThrNE, 256, 0, stream>>>(h1, N_NODES * D);

    // ---------------- Layer 2 (input h1 -> h2 = atom_embs, in d_out) ----------------
    fill_kernel<<<nThrN, 256, 0, stream>>>(mbuf, -INFINITY, N_NODES);
    fill_kernel<<<nThrN, 256, 0, stream>>>(den, 0.f, N_NODES);
    proj4_wmma<<<nTiles, 512, 0, stream>>>(h1, wq2, wk2, wv2, ws2,
                                           bq2, bk2, bv2, bs2, q, k, v, h2);
    edge_score_kernel<<<edgeBlk, 256, 0, stream>>>(q, k, ea, we2, src, dst, sc, mbuf);
    edge_exp_kernel<<<edgeThr, 256, 0, stream>>>(sc, mbuf, dst, den);
    edge_agg_kernel<<<edgeBlk, 256, 0, stream>>>(v, ea, we2, sc, den, src, dst, h2);
    leaky_kernel<<<nThrNE, 256, 0, stream>>>(h2, N_NODES * D);

    // ---------------- Pool + head ----------------
    fill_kernel<<<(N_GRAPH * D + 255) / 256, 256, 0, stream>>>(pooled, 0.f, N_GRAPH * D);
    pool_kernel<<<nThrNE, 256, 0, stream>>>(h2, batch, pooled);
    head_kernel<<<N_GRAPH, D, 0, stream>>>(pooled, wf1, bf1, wf2, bf2, outv);
}